// HomogeneousGCN_79474074845350
// MI455X (gfx1250) — compile-verified
//
#include <hip/hip_runtime.h>

typedef __attribute__((ext_vector_type(2))) float v2f;
typedef __attribute__((ext_vector_type(8))) float v8f;

#define EPS 1e-5f

// ---------------- utility ----------------
__global__ void k_zero(float* __restrict__ p, long long n) {
    long long i = (long long)blockIdx.x * blockDim.x + threadIdx.x;
    if (i < n) p[i] = 0.0f;
}

// ---------------- degree ----------------
__global__ void k_deg(const int* __restrict__ ei, float* __restrict__ deg, int E) {
    int e = blockIdx.x * blockDim.x + threadIdx.x;
    if (e < E) {
        int dst = ei[E + e];
        unsafeAtomicAdd(&deg[dst], 1.0f);
    }
}

__global__ void k_degfin(float* __restrict__ dis, float* __restrict__ dinv, int N) {
    int i = blockIdx.x * blockDim.x + threadIdx.x;
    if (i < N) {
        float d = dis[i] + 1.0f;   // self loop
        dis[i]  = rsqrtf(d);
        dinv[i] = 1.0f / d;
    }
}

// ---------------- WMMA f32 GEMM: C[M,Nc] = A[M,K] @ W[K,Nc] (+bias) ----------------
// One wave computes a 16-row tile, looping over Nc/16 column tiles and K/4 k-steps.
__global__ void k_gemm_wmma(const float* __restrict__ A, const float* __restrict__ W,
                            const float* __restrict__ bias, float* __restrict__ C,
                            int M, int K, int Nc) {
    const int lane = threadIdx.x & 31;
    const int wave = threadIdx.x >> 5;
    const int half = lane >> 4;      // 0: lanes 0-15, 1: lanes 16-31
    const int l    = lane & 15;
    const int tilesM = (M + 15) >> 4;
    int tm = blockIdx.x * (blockDim.x >> 5) + wave;
    if (tm >= tilesM) return;        // wave-uniform: EXEC stays all-ones for WMMA
    const int m0 = tm << 4;
    int rowA = m0 + l;
    if (rowA > M - 1) rowA = M - 1;  // clamped load (stores are guarded)
    const float* Arow = A + (long long)rowA * K;

    for (int n0 = 0; n0 < Nc; n0 += 16) {
        v8f c = {};
        for (int k0 = 0; k0 < K; k0 += 4) {
            const int ka = k0 + 2 * half;
            v2f a, b;
            a.x = Arow[ka];
            a.y = Arow[ka + 1];
            b.x = W[(long long)ka * Nc + n0 + l];
            b.y = W[(long long)(ka + 1) * Nc + n0 + l];
            c = __builtin_amdgcn_wmma_f32_16x16x4_f32(
                    false, a, false, b, (short)0, c, false, false);
        }
        float bv = bias ? bias[n0 + l] : 0.0f;
#pragma unroll
        for (int i = 0; i < 8; ++i) {
            int row = m0 + i + 8 * half;
            if (row < M) C[(long long)row * Nc + n0 + l] = c[i] + bv;
        }
    }
}

// ---------------- edge aggregation: agg[dst] += h[src] * dis[src]*dis[dst] ----------------
// 32 lanes per edge, one channel per lane: fully coalesced 128B gather + 128B atomic add.
__global__ void k_agg(const int* __restrict__ ei, const float* __restrict__ dis,
                      const float* __restrict__ h, float* __restrict__ agg, int E) {
    long long t = (long long)blockIdx.x * blockDim.x + threadIdx.x;
    long long e = t >> 5;
    int c = (int)(t & 31);
    if (e < E) {
        int src = ei[e];
        int dst = ei[E + e];
        float w = dis[src] * dis[dst];
        unsafeAtomicAdd(&agg[(long long)dst * 32 + c], h[(long long)src * 32 + c] * w);
    }
}

// ---------------- combine: y = agg + h * dinv[row] + b[c] ----------------
__global__ void k_combine(const float* __restrict__ agg, const float* __restrict__ h,
                          const float* __restrict__ dinv, const float* __restrict__ bias,
                          float* __restrict__ y, long long n32) {
    long long t = (long long)blockIdx.x * blockDim.x + threadIdx.x;
    if (t < n32) {
        int c = (int)(t & 31);
        long long r = t >> 5;
        y[t] = agg[t] + h[t] * dinv[r] + bias[c];
    }
}

// ---------------- per-channel sum / sumsq over rows (32 channels) ----------------
__global__ void k_stats(const float* __restrict__ y, long long n32, float* __restrict__ stats) {
    __shared__ float s0[256];
    __shared__ float s1[256];
    int tid = threadIdx.x;
    float sum = 0.0f, sq = 0.0f;
    // stride (gridDim*256) is a multiple of 32 -> each thread stays on channel tid&31
    for (long long t = (long long)blockIdx.x * 256 + tid; t < n32;
         t += (long long)gridDim.x * 256) {
        float v = y[t];
        sum += v;
        sq  += v * v;
    }
    s0[tid] = sum; s1[tid] = sq;
    __syncthreads();
    for (int s = 128; s >= 32; s >>= 1) {   // s multiple of 32: channel-preserving
        if (tid < s) { s0[tid] += s0[tid + s]; s1[tid] += s1[tid + s]; }
        __syncthreads();
    }
    if (tid < 32) {
        unsafeAtomicAdd(&stats[tid],      s0[tid]);
        unsafeAtomicAdd(&stats[32 + tid], s1[tid]);
    }
}

__global__ void k_bnfin(const float* __restrict__ stats, const float* __restrict__ gamma,
                        const float* __restrict__ beta, float* __restrict__ bnp, float invM) {
    int c = threadIdx.x;
    if (c < 32) {
        float mean = stats[c] * invM;
        float var  = stats[32 + c] * invM - mean * mean;
        float inv  = rsqrtf(var + EPS);
        float sc   = gamma[c] * inv;
        bnp[c]      = sc;
        bnp[32 + c] = beta[c] - mean * sc;
    }
}

__global__ void k_bnrelu(float* __restrict__ y, const float* __restrict__ bnp, long long n32) {
    long long t = (long long)blockIdx.x * blockDim.x + threadIdx.x;
    if (t < n32) {
        int c = (int)(t & 31);
        float v = y[t] * bnp[c] + bnp[32 + c];
        y[t] = v > 0.0f ? v : 0.0f;
    }
}

// ---------------- sum pooling over sorted batch ids ----------------
__global__ void k_pool(const float* __restrict__ z, const int* __restrict__ batch,
                       float* __restrict__ pooled, long long n32) {
    long long t = (long long)blockIdx.x * blockDim.x + threadIdx.x;
    if (t < n32) {
        int c = (int)(t & 31);
        long long r = t >> 5;
        int g = batch[r];
        unsafeAtomicAdd(&pooled[(long long)g * 32 + c], z[t]);
    }
}

// ---------------- final: out[g] = dot(h[g,:], M3) + mb3 ----------------
__global__ void k_mlp3(const float* __restrict__ h, const float* __restrict__ M3,
                       const float* __restrict__ mb3, float* __restrict__ out, int G) {
    int g = blockIdx.x * blockDim.x + threadIdx.x;
    if (g < G) {
        float s = mb3[0];
#pragma unroll
        for (int c = 0; c < 32; ++c) s += h[(long long)g * 32 + c] * M3[c];
        out[g] = s;
    }
}

// ---------------- host side ----------------
static inline int cdiv(long long a, long long b) { return (int)((a + b - 1) / b); }

extern "C" void kernel_launch(void* const* d_in, const int* in_sizes, int n_in,
                              void* d_out, int out_size, void* d_ws, size_t ws_size,
                              hipStream_t stream) {
    const float* x     = (const float*)d_in[0];
    const int*   ei    = (const int*)d_in[1];   // [2, E] int32
    const int*   batch = (const int*)d_in[2];   // [N] int32
    const float* W1 = (const float*)d_in[3];  const float* b1  = (const float*)d_in[4];
    const float* g1 = (const float*)d_in[5];  const float* be1 = (const float*)d_in[6];
    const float* W2 = (const float*)d_in[7];  const float* b2  = (const float*)d_in[8];
    const float* g2 = (const float*)d_in[9];  const float* be2 = (const float*)d_in[10];
    const float* W3 = (const float*)d_in[11]; const float* b3  = (const float*)d_in[12];
    const float* M1 = (const float*)d_in[13]; const float* mb1 = (const float*)d_in[14];
    const float* mg1= (const float*)d_in[15]; const float* mbe1= (const float*)d_in[16];
    const float* M2 = (const float*)d_in[17]; const float* mb2 = (const float*)d_in[18];
    const float* mg2= (const float*)d_in[19]; const float* mbe2= (const float*)d_in[20];
    const float* M3 = (const float*)d_in[21]; const float* mb3 = (const float*)d_in[22];

    const int N = in_sizes[0] / 128;   // 200000
    const int E = in_sizes[1] / 2;     // 6400000
    const int G = out_size;            // 1024 (OUT=1)

    // workspace layout (floats)
    float* ws   = (float*)d_ws;
    float* dis  = ws;                                  // N (degree, then D^-1/2)
    float* dinv = dis + N;                             // N
    float* hb   = dinv + N;                            // 32N  transformed features
    float* ab   = hb + (size_t)32 * N;                 // 32N  aggregation
    float* zb   = ab + (size_t)32 * N;                 // 32N  layer output
    float* pooled = zb + (size_t)32 * N;               // 32G
    float* gb1  = pooled + (size_t)32 * G;             // 32G
    float* gb2  = gb1 + (size_t)32 * G;                // 32G
    float* stats= gb2 + (size_t)32 * G;                // 64
    float* bnp  = stats + 64;                          // 64 (scale|shift)

    const long long n32 = (long long)N * 32;
    const long long g32 = (long long)G * 32;

    // degrees + norms
    k_zero<<<cdiv(N, 256), 256, 0, stream>>>(dis, N);
    k_deg<<<cdiv(E, 256), 256, 0, stream>>>(ei, dis, E);
    k_degfin<<<cdiv(N, 256), 256, 0, stream>>>(dis, dinv, N);

    // one GCN conv (+ optional BN/ReLU). in: [N,K] -> zb: [N,32]
    auto conv = [&](const float* in, int K, const float* W, const float* b,
                    const float* gam, const float* bet, bool bn) {
        int gemmBlocks = cdiv((N + 15) / 16, 8);
        k_gemm_wmma<<<gemmBlocks, 256, 0, stream>>>(in, W, nullptr, hb, N, K, 32);
        k_zero<<<cdiv(n32, 256), 256, 0, stream>>>(ab, n32);
        k_agg<<<cdiv((long long)E * 32, 256), 256, 0, stream>>>(ei, dis, hb, ab, E);
        k_combine<<<cdiv(n32, 256), 256, 0, stream>>>(ab, hb, dinv, b, zb, n32);
        if (bn) {
            k_zero<<<1, 64, 0, stream>>>(stats, 64);
            k_stats<<<1024, 256, 0, stream>>>(zb, n32, stats);
            k_bnfin<<<1, 32, 0, stream>>>(stats, gam, bet, bnp, 1.0f / (float)N);
            k_bnrelu<<<cdiv(n32, 256), 256, 0, stream>>>(zb, bnp, n32);
        }
    };

    conv(x,  128, W1, b1, g1, be1, true);
    conv(zb,  32, W2, b2, g2, be2, true);
    conv(zb,  32, W3, b3, nullptr, nullptr, false);

    // sum pooling
    k_zero<<<cdiv(g32, 256), 256, 0, stream>>>(pooled, g32);
    k_pool<<<cdiv(n32, 256), 256, 0, stream>>>(zb, batch, pooled, n32);

    // MLP layer (WMMA) + BN + ReLU
    auto mlp = [&](const float* in, const float* W, const float* b, float* out,
                   const float* gam, const float* bet) {
        int gemmBlocks = cdiv((G + 15) / 16, 8);
        k_gemm_wmma<<<gemmBlocks, 256, 0, stream>>>(in, W, b, out, G, 32, 32);
        k_zero<<<1, 64, 0, stream>>>(stats, 64);
        k_stats<<<128, 256, 0, stream>>>(out, g32, stats);
        k_bnfin<<<1, 32, 0, stream>>>(stats, gam, bet, bnp, 1.0f / (float)G);
        k_bnrelu<<<cdiv(g32, 256), 256, 0, stream>>>(out, bnp, g32);
    };

    mlp(pooled, M1, mb1, gb1, mg1, mbe1);
    mlp(gb1,    M2, mb2, gb2, mg2, mbe2);

    k_mlp3<<<cdiv(G, 256), 256, 0, stream>>>(gb2, M3, mb3, (float*)d_out, G);
}